// Classifier_GCN_1176821039655
// MI455X (gfx1250) — compile-verified
//
#include <hip/hip_runtime.h>
#include <hip/hip_bf16.h>
#include <math.h>

#define N_NODES 100000
#define HID     128
#define NUM_CLS 20
#define N_EDGES 625000
#define CLS_PAD 32

typedef __attribute__((ext_vector_type(2))) float v2f;
typedef __attribute__((ext_vector_type(8))) float v8f;

// ---------------------------------------------------------------------------
// Edge scatter: mean-aggregation numerator + counts via global f32 atomics.
// 32 threads per edge, 4 channels per thread. Grid covers E*32 exactly.
// ---------------------------------------------------------------------------
__global__ void edge_scatter(const float* __restrict__ x,
                             const long long* __restrict__ ei,
                             float* __restrict__ summed,
                             float* __restrict__ counts) {
    int t = blockIdx.x * blockDim.x + threadIdx.x;   // < N_EDGES*32
    int e = t >> 5;
    int q = (t & 31) << 2;
    int s = (int)ei[e];
    int d = (int)ei[N_EDGES + e];
    const float* xr = x + (size_t)s * HID + q;
    float* sr = summed + (size_t)d * HID + q;
    atomicAdd(sr + 0, xr[0]);
    atomicAdd(sr + 1, xr[1]);
    atomicAdd(sr + 2, xr[2]);
    atomicAdd(sr + 3, xr[3]);
    if ((t & 31) == 0) atomicAdd(counts + d, 1.0f);
}

// ---------------------------------------------------------------------------
// Column-normalize W_cls [128][20] into zero-padded Wn [128][32].
// One block, 128 threads.
// ---------------------------------------------------------------------------
__global__ void norm_wcls(const float* __restrict__ Wcls,
                          float* __restrict__ Wn) {
    __shared__ float inv[CLS_PAD];
    int t = threadIdx.x;
    if (t < CLS_PAD) {
        float s = 0.0f;
        if (t < NUM_CLS) {
            for (int k = 0; k < HID; ++k) {
                float w = Wcls[k * NUM_CLS + t];
                s += w * w;
            }
        }
        inv[t] = (t < NUM_CLS) ? (1.0f / fmaxf(sqrtf(s), 1e-12f)) : 0.0f;
    }
    __syncthreads();
    int k = t;                                  // 128 threads = 128 rows
    for (int c = 0; c < CLS_PAD; ++c)
        Wn[k * CLS_PAD + c] = (c < NUM_CLS) ? (Wcls[k * NUM_CLS + c] * inv[c])
                                            : 0.0f;
}

// ---------------------------------------------------------------------------
// Fused: aggr = summed/max(cnt,1);  h = [aggr|x] @ [W_l;W_r] + b_l;
//        out = l2norm_rows(h) @ Wn   — all via V_WMMA_F32_16X16X4_F32.
// One wave (32 lanes) per 16-node tile; 2 waves per block.
// ---------------------------------------------------------------------------
__global__ __launch_bounds__(64) void sage_fused(
    const float* __restrict__ x,
    const float* __restrict__ summed,
    const float* __restrict__ counts,
    const float* __restrict__ W_l,
    const float* __restrict__ b_l,
    const float* __restrict__ W_r,
    const float* __restrict__ Wn,
    float* __restrict__ out) {

    __shared__ float lds[2 * 16 * 256];          // 32 KB: 16x256 tile per wave
    const int wave = threadIdx.x >> 5;
    const int lane = threadIdx.x & 31;
    const int tile = blockIdx.x * 2 + wave;      // 0..6249 (exact)
    const int m0   = tile * 16;
    float* T = lds + wave * (16 * 256);

    // ---- Stage A-tile into LDS: cols 0..127 = aggr (mean), 128..255 = x ----
    for (int i = 0; i < 64; ++i) {
        int idx = i * 32 + lane;                 // 0..2047, coalesced
        int row = idx >> 7;
        int c   = idx & 127;
        float cnt = counts[m0 + row];
        float invc = 1.0f / fmaxf(cnt, 1.0f);
        T[row * 256 + c]       = summed[(size_t)(m0 + row) * HID + c] * invc;
        T[row * 256 + 128 + c] = x[(size_t)(m0 + row) * HID + c];
    }
    __syncthreads();

    const int dk = (lane < 16) ? 0 : 2;          // A/B fragment K sub-offset
    const int rr = lane & 15;                    // M (for A) / N (for B,C,D)

    // ---- Accumulators preloaded with bias (bias depends only on column) ----
    v8f acc[8];
#pragma unroll
    for (int nt = 0; nt < 8; ++nt) {
        float bl = b_l[nt * 16 + rr];
#pragma unroll
        for (int r = 0; r < 8; ++r) acc[nt][r] = bl;
    }

    // ---- Fused K=256 GEMM: k<128 -> aggr@W_l ; k>=128 -> x@W_r ----
    for (int kk = 0; kk < 256; kk += 4) {
        v2f a;
        a.x = T[rr * 256 + kk + dk];
        a.y = T[rr * 256 + kk + dk + 1];
        const float* W = (kk < 128) ? (W_l + (size_t)kk * HID)
                                    : (W_r + (size_t)(kk - 128) * HID);
#pragma unroll
        for (int nt = 0; nt < 8; ++nt) {
            v2f b;
            b.x = W[(size_t)dk * HID + nt * 16 + rr];
            b.y = W[(size_t)(dk + 1) * HID + nt * 16 + rr];
            acc[nt] = __builtin_amdgcn_wmma_f32_16x16x4_f32(
                false, a, false, b, (short)0, acc[nt], false, false);
        }
    }

    // ---- Spill h tile (D-layout) into LDS as [16][128] (reuse tile mem) ----
#pragma unroll
    for (int nt = 0; nt < 8; ++nt) {
#pragma unroll
        for (int r = 0; r < 8; ++r) {
            int m = r + ((lane < 16) ? 0 : 8);
            T[m * 128 + nt * 16 + rr] = acc[nt][r];
        }
    }
    __syncthreads();

    // ---- Row L2 norm: lane handles row rr, half the columns ----
    {
        int cb = (lane < 16) ? 0 : 64;
        float s = 0.0f;
        for (int j = 0; j < 64; ++j) {
            float v = T[rr * 128 + cb + j];
            s += v * v;
        }
        s += __shfl_xor(s, 16, 32);              // combine the two halves
        float invn = 1.0f / fmaxf(sqrtf(s), 1e-12f);
        for (int j = 0; j < 64; ++j)
            T[rr * 128 + cb + j] *= invn;
    }
    __syncthreads();

    // ---- Classifier GEMM: h_norm (16x128) @ Wn (128x32, cols>=20 zero) ----
    v8f c0, c1;
#pragma unroll
    for (int r = 0; r < 8; ++r) { c0[r] = 0.0f; c1[r] = 0.0f; }

    for (int kk = 0; kk < 128; kk += 4) {
        v2f a;
        a.x = T[rr * 128 + kk + dk];
        a.y = T[rr * 128 + kk + dk + 1];
        v2f b0, b1;
        b0.x = Wn[(kk + dk) * CLS_PAD + rr];
        b0.y = Wn[(kk + dk + 1) * CLS_PAD + rr];
        b1.x = Wn[(kk + dk) * CLS_PAD + 16 + rr];
        b1.y = Wn[(kk + dk + 1) * CLS_PAD + 16 + rr];
        c0 = __builtin_amdgcn_wmma_f32_16x16x4_f32(
            false, a, false, b0, (short)0, c0, false, false);
        c1 = __builtin_amdgcn_wmma_f32_16x16x4_f32(
            false, a, false, b1, (short)0, c1, false, false);
    }

    // ---- Write the 20 valid class columns ----
#pragma unroll
    for (int r = 0; r < 8; ++r) {
        int m = m0 + r + ((lane < 16) ? 0 : 8);
        out[(size_t)m * NUM_CLS + rr] = c0[r];
        if (rr < 4) out[(size_t)m * NUM_CLS + 16 + rr] = c1[r];
    }
}

// ---------------------------------------------------------------------------
extern "C" void kernel_launch(void* const* d_in, const int* in_sizes, int n_in,
                              void* d_out, int out_size, void* d_ws, size_t ws_size,
                              hipStream_t stream) {
    (void)in_sizes; (void)n_in; (void)out_size; (void)ws_size;

    const float*     x    = (const float*)d_in[0];
    const long long* ei   = (const long long*)d_in[1];   // int64 edge_index
    const float*     W_l  = (const float*)d_in[2];
    const float*     b_l  = (const float*)d_in[3];
    const float*     W_r  = (const float*)d_in[4];
    const float*     Wcls = (const float*)d_in[5];
    float*           out  = (float*)d_out;

    float* summed = (float*)d_ws;                        // [N_NODES][HID]
    float* counts = summed + (size_t)N_NODES * HID;      // [N_NODES]
    float* Wn     = counts + N_NODES;                    // [HID][CLS_PAD]

    // Zero accumulators every call (graph-replay safe).
    hipMemsetAsync(d_ws, 0,
                   ((size_t)N_NODES * HID + N_NODES) * sizeof(float), stream);

    edge_scatter<<<(N_EDGES * 32) / 256, 256, 0, stream>>>(x, ei, summed, counts);
    norm_wcls<<<1, 128, 0, stream>>>(Wcls, Wn);
    sage_fused<<<N_NODES / 32, 64, 0, stream>>>(x, summed, counts,
                                                W_l, b_l, W_r, Wn, out);
}